// GraphFeaturesExtractor_46411416600839
// MI455X (gfx1250) — compile-verified
//
#include <hip/hip_runtime.h>
#include <hip/hip_bf16.h>

typedef float v2f __attribute__((ext_vector_type(2)));
typedef float v8f __attribute__((ext_vector_type(8)));

#define NEG_SLOPE 0.2f

// Order-preserving encoding of f32 into u32 so atomicMax(u32) == float max.
__device__ __forceinline__ unsigned enc_f32(float f) {
  unsigned u = __float_as_uint(f);
  return (u & 0x80000000u) ? ~u : (u | 0x80000000u);
}
__device__ __forceinline__ float dec_f32(unsigned e) {
  return (e & 0x80000000u) ? __uint_as_float(e & 0x7FFFFFFFu)
                           : __uint_as_float(~e);
}

// ---------------------------------------------------------------------------
// C[M,Nc] = A[M,K] @ B[K,Nc] in fp32 via V_WMMA_F32_16X16X4_F32.
// One wave per 16x16 output tile. blockDim.x = 128 (4 waves -> 4 N-tiles).
// Optionally fuses bias add + ReLU (for the FC head).
// A-matrix layout (16x4 f32): lane<16 -> row=lane, K={k0,k0+1};
//                             lane>=16 -> row=lane-16, K={k0+2,k0+3}.
// B mirrors A transposed; C/D: VGPR j -> row = j + 8*(lane>=16), col = lane&15.
// ---------------------------------------------------------------------------
__global__ void gemm16_wmma_f32(const float* __restrict__ A,
                                const float* __restrict__ B,
                                const float* __restrict__ bias,
                                float* __restrict__ C,
                                int M, int K, int Nc, int fuse_bias_relu) {
  const int lane = threadIdx.x & 31;
  const int wave = threadIdx.x >> 5;
  const int m0 = blockIdx.x << 4;
  const int n0 = (blockIdx.y * (blockDim.x >> 5) + wave) << 4;
  if (n0 >= Nc) return;  // wave-uniform exit, EXEC stays all-ones for WMMA
  const int half = lane >> 4;
  const int l = lane & 15;
  const int arow = min(m0 + l, M - 1);

  v8f acc = {};
  const float* ap = A + (size_t)arow * K + 2 * half;
  const float* bp = B + (size_t)(2 * half) * Nc + (n0 + l);
  for (int k0 = 0; k0 < K; k0 += 4) {
    v2f a, b;
    a.x = ap[0];
    a.y = ap[1];
    b.x = bp[0];
    b.y = bp[Nc];
    acc = __builtin_amdgcn_wmma_f32_16x16x4_f32(
        /*neg_a=*/false, a, /*neg_b=*/false, b,
        /*c_mod=*/(short)0, acc, /*reuse_a=*/false, /*reuse_b=*/false);
    ap += 4;
    bp += (size_t)4 * Nc;
  }

  const int col = n0 + l;
  const float bval = fuse_bias_relu ? bias[col] : 0.0f;
#pragma unroll
  for (int j = 0; j < 8; ++j) {
    int row = m0 + j + 8 * half;
    if (row < M) {
      float v = acc[j];
      if (fuse_bias_relu) v = fmaxf(v + bval, 0.0f);
      C[(size_t)row * Nc + col] = v;
    }
  }
}

// Append self-loops: srcL/dstL = [edge_index[0] ; 0..N-1], [edge_index[1] ; 0..N-1]
__global__ void build_edges_k(const int* __restrict__ ei, int* __restrict__ s,
                              int* __restrict__ d, int E, int N) {
  int t = blockIdx.x * blockDim.x + threadIdx.x;
  if (t < E) {
    s[t] = ei[t];
    d[t] = ei[E + t];
  } else if (t < E + N) {
    int n = t - E;
    s[t] = n;
    d[t] = n;
  }
}

// alpha_s[n,h] = sum_c h[n,h,c]*a_src[h,c]; alpha_d likewise.
__global__ void alpha_k(const float* __restrict__ h, const float* __restrict__ a_s,
                        const float* __restrict__ a_d, float* __restrict__ als,
                        float* __restrict__ ald, int N, int H, int C) {
  int t = blockIdx.x * blockDim.x + threadIdx.x;  // t = n*H + hh
  if (t >= N * H) return;
  int hh = t % H;
  const float* hp = h + (size_t)t * C;  // h[n*H*C + hh*C] == h[t*C]
  float ss = 0.f, sd = 0.f;
  for (int c = 0; c < C; ++c) {
    float v = hp[c];
    ss = fmaf(v, a_s[hh * C + c], ss);
    sd = fmaf(v, a_d[hh * C + c], sd);
  }
  als[t] = ss;
  ald[t] = sd;
}

// e = leaky_relu(alpha_s[src]+alpha_d[dst]); segment max over dst via u32 atomicMax.
__global__ void edge_pass1_k(const int* __restrict__ s, const int* __restrict__ d,
                             const float* __restrict__ als, const float* __restrict__ ald,
                             float* __restrict__ e, unsigned* __restrict__ menc,
                             int Et, int H) {
  int t = blockIdx.x * blockDim.x + threadIdx.x;
  if (t >= Et * H) return;
  int k = t / H, hh = t - k * H;
  int si = s[k], di = d[k];
  float v = als[si * H + hh] + ald[di * H + hh];
  v = v > 0.f ? v : NEG_SLOPE * v;
  e[t] = v;
  atomicMax(&menc[di * H + hh], enc_f32(v));
}

// e <- exp(e - m[dst]); denom[dst] += e
__global__ void edge_pass2_k(const int* __restrict__ d, float* __restrict__ e,
                             const unsigned* __restrict__ menc,
                             float* __restrict__ denom, int Et, int H) {
  int t = blockIdx.x * blockDim.x + threadIdx.x;
  if (t >= Et * H) return;
  int k = t / H, hh = t - k * H;
  int di = d[k];
  float v = __expf(e[t] - dec_f32(menc[di * H + hh]));
  e[t] = v;
  atomicAdd(&denom[di * H + hh], v);
}

// e <- e / denom[dst]   (normalized attention coefficient)
__global__ void edge_coeff_k(const int* __restrict__ d, float* __restrict__ e,
                             const float* __restrict__ denom, int Et, int H) {
  int t = blockIdx.x * blockDim.x + threadIdx.x;
  if (t >= Et * H) return;
  int k = t / H, hh = t - k * H;
  e[t] = e[t] / denom[d[k] * H + hh];
}

// agg[dst, ch..ch+3] += coeff * h[src, ch..ch+3]  (float4 gather, 4 f32 atomics)
__global__ void edge_agg_k(const int* __restrict__ s, const int* __restrict__ d,
                           const float* __restrict__ coeff, const float* __restrict__ h,
                           float* __restrict__ agg, int Et, int H, int C) {
  const int HC = H * C;
  const int vec = HC >> 2;
  int t = blockIdx.x * blockDim.x + threadIdx.x;
  if (t >= Et * vec) return;
  int k = t / vec, q = t - k * vec;
  int ch = q << 2;
  int hh = ch / C;
  int si = s[k], di = d[k];
  float w = coeff[k * H + hh];
  const float4 hv = *(const float4*)(h + (size_t)si * HC + ch);
  float* ap = agg + (size_t)di * HC + ch;
  atomicAdd(ap + 0, w * hv.x);
  atomicAdd(ap + 1, w * hv.y);
  atomicAdd(ap + 2, w * hv.z);
  atomicAdd(ap + 3, w * hv.w);
}

// x <- relu(x + bias[ch])
__global__ void bias_relu_k(float* __restrict__ x, const float* __restrict__ b,
                            int total, int HC) {
  int t = blockIdx.x * blockDim.x + threadIdx.x;
  if (t >= total) return;
  x[t] = fmaxf(x[t] + b[t % HC], 0.f);
}

extern "C" void kernel_launch(void* const* d_in, const int* in_sizes, int n_in,
                              void* d_out, int out_size, void* d_ws, size_t ws_size,
                              hipStream_t stream) {
  const float* x = (const float*)d_in[0];
  const int* ei = (const int*)d_in[1];
  const float* W[3]  = {(const float*)d_in[2], (const float*)d_in[6],  (const float*)d_in[10]};
  const float* AS[3] = {(const float*)d_in[3], (const float*)d_in[7],  (const float*)d_in[11]};
  const float* AD[3] = {(const float*)d_in[4], (const float*)d_in[8],  (const float*)d_in[12]};
  const float* BI[3] = {(const float*)d_in[5], (const float*)d_in[9],  (const float*)d_in[13]};
  const float* fcw = (const float*)d_in[14];
  const float* fcb = (const float*)d_in[15];

  const int N = in_sizes[0] / 128;
  const int E = in_sizes[1] / 2;
  const int Et = E + N;

  // Carve scratch out of d_ws (256B-aligned slabs).
  char* p = (char*)d_ws;
  auto grab = [&](size_t bytes) -> void* {
    void* r = (void*)p;
    p += (bytes + 255) & ~(size_t)255;
    return r;
  };
  int*      srcL  = (int*)grab((size_t)Et * 4);
  int*      dstL  = (int*)grab((size_t)Et * 4);
  float*    hbuf  = (float*)grab((size_t)N * 128 * 4);
  float*    featA = (float*)grab((size_t)N * 128 * 4);
  float*    featB = (float*)grab((size_t)N * 128 * 4);
  float*    alS   = (float*)grab((size_t)N * 2 * 4);
  float*    alD   = (float*)grab((size_t)N * 2 * 4);
  unsigned* menc  = (unsigned*)grab((size_t)N * 2 * 4);
  float*    den   = (float*)grab((size_t)N * 2 * 4);
  float*    ecf   = (float*)grab((size_t)Et * 2 * 4);

  const int TB = 256;
  build_edges_k<<<(Et + TB - 1) / TB, TB, 0, stream>>>(ei, srcL, dstL, E, N);

  auto gat_layer = [&](const float* X, int Fin, int li, int H, int C, float* out) {
    const int HC = H * C;
    // h = X @ W   (fp32 WMMA)
    dim3 gg((unsigned)((N + 15) / 16), (unsigned)((HC + 63) / 64));
    gemm16_wmma_f32<<<gg, 128, 0, stream>>>(X, W[li], nullptr, hbuf, N, Fin, HC, 0);
    alpha_k<<<(N * H + TB - 1) / TB, TB, 0, stream>>>(hbuf, AS[li], AD[li], alS, alD, N, H, C);
    hipMemsetAsync(menc, 0, (size_t)N * H * 4, stream);  // enc(any real) > 0
    hipMemsetAsync(den, 0, (size_t)N * H * 4, stream);
    hipMemsetAsync(out, 0, (size_t)N * HC * 4, stream);
    const int eh = Et * H;
    edge_pass1_k<<<(eh + TB - 1) / TB, TB, 0, stream>>>(srcL, dstL, alS, alD, ecf, menc, Et, H);
    edge_pass2_k<<<(eh + TB - 1) / TB, TB, 0, stream>>>(dstL, ecf, menc, den, Et, H);
    edge_coeff_k<<<(eh + TB - 1) / TB, TB, 0, stream>>>(dstL, ecf, den, Et, H);
    const int ev = Et * (HC >> 2);
    edge_agg_k<<<(ev + TB - 1) / TB, TB, 0, stream>>>(srcL, dstL, ecf, hbuf, out, Et, H, C);
    bias_relu_k<<<(N * HC + TB - 1) / TB, TB, 0, stream>>>(out, BI[li], N * HC, HC);
  };

  gat_layer(x,     128, 0, 2, 32, featA);  // -> featA [N,64]
  gat_layer(featA,  64, 1, 2, 64, featB);  // -> featB [N,128]
  gat_layer(featB, 128, 2, 1, 64, featA);  // -> featA [N,64]

  // FC head: relu(featA @ fcw + fcb) -> d_out [N,512]
  dim3 gf((unsigned)((N + 15) / 16), (unsigned)((512 + 63) / 64));
  gemm16_wmma_f32<<<gf, 128, 0, stream>>>(featA, fcw, fcb, (float*)d_out, N, 64, 512, 1);
}